// GMM_10797547782831
// MI455X (gfx1250) — compile-verified
//
#include <hip/hip_runtime.h>
#include <math.h>

#define KCOMP 16
#define DDIM  8
#define LOG_2PI  1.8378770664093453f   // ln(2*pi)
#define LOG2E_F  1.4426950408889634f   // log2(e)
#define LN2_D    0.6931471805599453    // ln(2)

typedef float v2f __attribute__((ext_vector_type(2)));
typedef float v8f __attribute__((ext_vector_type(8)));

struct GmmWs {
  double acc;          // global f64 accumulator (log2-domain sum, counted twice)
  double pad;
  float  W[16 * 16];   // W[f*16 + k] * log2e : feature f (0..7 lin, 8..15 quad), comp k
  float  cst[16];      // per-component constant (incl. log p), * log2e
};

// ---------------- prep: build log2-domain W and constants, zero accumulator ----------------
__global__ void gmm_prep(const float* __restrict__ theta, GmmWs* __restrict__ ws) {
  int k = threadIdx.x;
  if (k == 0) ws->acc = 0.0;
  if (k < KCOMP) {
    float p = theta[k];
    float csum = logf(p) - 0.5f * (float)DDIM * LOG_2PI;
    for (int d = 0; d < DDIM; ++d) {
      float mu = theta[KCOMP + k * DDIM + d];
      float sg = theta[KCOMP + KCOMP * DDIM + k * DDIM + d];
      float inv2 = 1.0f / (sg * sg);
      ws->W[d * 16 + k]       = LOG2E_F * (mu * inv2);      // coeff of x_d
      ws->W[(8 + d) * 16 + k] = LOG2E_F * (-0.5f * inv2);   // coeff of x_d^2
      csum += -0.5f * mu * mu * inv2 - logf(sg);
    }
    ws->cst[k] = LOG2E_F * csum;
  }
}

// ---------------- main: streaming WMMA GEMM (components x points), log2-domain LSE --------
// D = W^T (16comp x 16feat) * F (16feat x 16pts); two independent 16-pt tiles per iteration.
// D layout: VGPR j, lanes 0-15 -> comp j, point lane; lanes 16-31 -> comp j+8, point lane-16.
// logsumexp(point) = 7 reg-max + 1 xor16, 8 exp2 + 7 add + 1 xor16, + log2. Both halves keep
// the (identical) result; final sum is halved once.
__global__ void gmm_main(const float* __restrict__ x, const GmmWs* __restrict__ ws,
                         long long n, double* __restrict__ acc_out) {
  const int lane = threadIdx.x & 31;
  const int hv   = lane >> 4;
  const int lm   = lane & 15;

  // Constant A tiles (W^T): a_t VGPR0 = f=4t+2hv, VGPR1 = f=4t+1+2hv, row m = lm
  v2f a0, a1, a2, a3;
  a0.x = ws->W[(0  + 2 * hv) * 16 + lm];  a0.y = ws->W[(1  + 2 * hv) * 16 + lm];
  a1.x = ws->W[(4  + 2 * hv) * 16 + lm];  a1.y = ws->W[(5  + 2 * hv) * 16 + lm];
  a2.x = ws->W[(8  + 2 * hv) * 16 + lm];  a2.y = ws->W[(9  + 2 * hv) * 16 + lm];
  a3.x = ws->W[(12 + 2 * hv) * 16 + lm];  a3.y = ws->W[(13 + 2 * hv) * 16 + lm];

  float cinit[8];
  #pragma unroll
  for (int j = 0; j < 8; ++j) cinit[j] = ws->cst[8 * hv + j];

  const long long pairs = n >> 5;                         // 32-point macro-tiles (no tail here)
  const long long wid = (long long)blockIdx.x * (blockDim.x >> 5) + (threadIdx.x >> 5);
  const long long nw  = (long long)gridDim.x * (blockDim.x >> 5);

  const float* xr = x + ((wid << 5) + lm) * DDIM;         // lane's point in tile 0 of pair
  const long long step = (nw << 5) * DDIM;                // floats to advance per iteration

  double local = 0.0;

  for (long long p = wid; p < pairs; p += nw, xr += step) {
    // ---- tile 0 (points base..base+15) ----
    v2f b0 = *(const v2f*)(xr + 2 * hv);
    v2f b1 = *(const v2f*)(xr + 4 + 2 * hv);
    // ---- tile 1 (points base+16..base+31) ----
    v2f d0 = *(const v2f*)(xr + 128 + 2 * hv);
    v2f d1 = *(const v2f*)(xr + 132 + 2 * hv);

    v2f b2; b2.x = b0.x * b0.x; b2.y = b0.y * b0.y;
    v2f b3; b3.x = b1.x * b1.x; b3.y = b1.y * b1.y;
    v2f d2; d2.x = d0.x * d0.x; d2.y = d0.y * d0.y;
    v2f d3; d3.x = d1.x * d1.x; d3.y = d1.y * d1.y;

    v8f c = { cinit[0], cinit[1], cinit[2], cinit[3],
              cinit[4], cinit[5], cinit[6], cinit[7] };
    v8f e = c;
    c = __builtin_amdgcn_wmma_f32_16x16x4_f32(false, a0, false, b0, (short)0, c, false, false);
    e = __builtin_amdgcn_wmma_f32_16x16x4_f32(false, a0, false, d0, (short)0, e, false, false);
    c = __builtin_amdgcn_wmma_f32_16x16x4_f32(false, a1, false, b1, (short)0, c, false, false);
    e = __builtin_amdgcn_wmma_f32_16x16x4_f32(false, a1, false, d1, (short)0, e, false, false);
    c = __builtin_amdgcn_wmma_f32_16x16x4_f32(false, a2, false, b2, (short)0, c, false, false);
    e = __builtin_amdgcn_wmma_f32_16x16x4_f32(false, a2, false, d2, (short)0, e, false, false);
    c = __builtin_amdgcn_wmma_f32_16x16x4_f32(false, a3, false, b3, (short)0, c, false, false);
    e = __builtin_amdgcn_wmma_f32_16x16x4_f32(false, a3, false, d3, (short)0, e, false, false);

    // ---- epilogue tile 0 ----
    float m0 = c[0];
    #pragma unroll
    for (int j = 1; j < 8; ++j) m0 = fmaxf(m0, c[j]);
    m0 = fmaxf(m0, __shfl_xor(m0, 16, 32));
    float s0 = 0.0f;
    #pragma unroll
    for (int j = 0; j < 8; ++j) s0 += __builtin_amdgcn_exp2f(c[j] - m0);
    s0 += __shfl_xor(s0, 16, 32);
    float l0 = m0 + __builtin_amdgcn_logf(s0);     // log2-domain lse

    // ---- epilogue tile 1 ----
    float m1 = e[0];
    #pragma unroll
    for (int j = 1; j < 8; ++j) m1 = fmaxf(m1, e[j]);
    m1 = fmaxf(m1, __shfl_xor(m1, 16, 32));
    float s1 = 0.0f;
    #pragma unroll
    for (int j = 0; j < 8; ++j) s1 += __builtin_amdgcn_exp2f(e[j] - m1);
    s1 += __shfl_xor(s1, 16, 32);
    float l1 = m1 + __builtin_amdgcn_logf(s1);

    local += (double)(l0 + l1);                    // both halves accumulate; halved at the end
  }

  local += __shfl_xor(local, 16, 32);
  local += __shfl_xor(local, 8, 32);
  local += __shfl_xor(local, 4, 32);
  local += __shfl_xor(local, 2, 32);
  local += __shfl_xor(local, 1, 32);
  if (lane == 0) atomicAdd(acc_out, local);
}

// ---------------- finalize: tail rows (n mod 32), scale, emit float --------------------------
__global__ void gmm_fin(const double* __restrict__ acc, const float* __restrict__ x,
                        const float* __restrict__ theta, const int* __restrict__ fds,
                        long long n, float* __restrict__ out) {
  if (threadIdx.x == 0 && blockIdx.x == 0) {
    double tail = 0.0;
    const long long start = (n >> 5) << 5;
    for (long long r = start; r < n; ++r) {
      double vmax = -1.0e300;
      double v[KCOMP];
      for (int k = 0; k < KCOMP; ++k) {
        double lc = log((double)theta[k]) - 0.5 * (double)DDIM * (double)LOG_2PI;
        for (int d = 0; d < DDIM; ++d) {
          double mu = theta[KCOMP + k * DDIM + d];
          double sg = theta[KCOMP + KCOMP * DDIM + k * DDIM + d];
          double z  = ((double)x[r * DDIM + d] - mu) / sg;
          lc += -0.5 * z * z - log(sg);
        }
        v[k] = lc;
        vmax = vmax > lc ? vmax : lc;
      }
      double s = 0.0;
      for (int k = 0; k < KCOMP; ++k) s += exp(v[k] - vmax);
      tail += vmax + log(s);
    }
    double total = acc[0] * 0.5 * LN2_D + tail;    // halve double-count, log2 -> ln
    out[0] = (float)(total * ((double)fds[0] / (double)n));
  }
}

extern "C" void kernel_launch(void* const* d_in, const int* in_sizes, int n_in,
                              void* d_out, int out_size, void* d_ws, size_t ws_size,
                              hipStream_t stream) {
  const float* x     = (const float*)d_in[0];
  const float* theta = (const float*)d_in[1];
  const int*   fds   = (const int*)d_in[2];
  const long long n  = (long long)in_sizes[0] / DDIM;

  GmmWs* ws = (GmmWs*)d_ws;

  hipLaunchKernelGGL(gmm_prep, dim3(1), dim3(32), 0, stream, theta, ws);

  const int threads = 256;                      // 8 waves per block
  const int blocks  = 1024;                     // grid-stride over 62.5k pair-tiles
  hipLaunchKernelGGL(gmm_main, dim3(blocks), dim3(threads), 0, stream,
                     x, ws, n, &ws->acc);

  hipLaunchKernelGGL(gmm_fin, dim3(1), dim3(1), 0, stream,
                     &ws->acc, x, theta, fds, n, (float*)d_out);
}